// CFTransformerEncoderVN_45646912422491
// MI455X (gfx1250) — compile-verified
//
#include <hip/hip_runtime.h>
#include <math.h>

#define NS_ 256
#define NV_ 64
#define NN_c 15000
#define EE_c 480000
#define L5 5

typedef float v2f __attribute__((ext_vector_type(2)));
typedef float v8f __attribute__((ext_vector_type(8)));

static __device__ __forceinline__ v8f wmma_f32(v2f a, v2f b, v8f c) {
  return __builtin_amdgcn_wmma_f32_16x16x4_f32(false, a, false, b, (short)0, c, false, false);
}

// One 16x16 output tile of  Out[e][n0+j] += sum_k A[e][k] * W[n0+j][k]
// A: LDS row-major [16][lda]; W: global row-major (outDim x K), ldw = K stride.
static __device__ __forceinline__ v8f gemm_lds(const float* A, int lda,
                                               const float* W, int ldw,
                                               int n0, int K, v8f acc) {
  const int lane = threadIdx.x & 31;
  const int l15  = lane & 15;
  const int koff = (lane >> 4) << 1;
  const float* arow = A + l15 * lda;
  const float* wrow = W + (size_t)(n0 + l15) * ldw;
  for (int k = 0; k < K; k += 4) {
    v2f a, b;
    a.x = arow[k + koff];     a.y = arow[k + koff + 1];
    b.x = wrow[k + koff];     b.y = wrow[k + koff + 1];
    acc = wmma_f32(a, b, acc);
  }
  return acc;
}

// K = 64 (A0, lda=64) concat 256 (A1, lda=256), W is (outDim x 320)
static __device__ __forceinline__ v8f gemm_cat320(const float* A0, const float* A1,
                                                  const float* W, int n0, v8f acc) {
  const int lane = threadIdx.x & 31;
  const int l15  = lane & 15;
  const int koff = (lane >> 4) << 1;
  const float* a0 = A0 + l15 * NV_;
  const float* a1 = A1 + l15 * NS_;
  const float* wrow = W + (size_t)(n0 + l15) * 320;
  for (int k = 0; k < 64; k += 4) {
    v2f a, b;
    a.x = a0[k + koff];        a.y = a0[k + koff + 1];
    b.x = wrow[k + koff];      b.y = wrow[k + koff + 1];
    acc = wmma_f32(a, b, acc);
  }
  for (int k = 0; k < 256; k += 4) {
    v2f a, b;
    a.x = a1[k + koff];          a.y = a1[k + koff + 1];
    b.x = wrow[64 + k + koff];   b.y = wrow[64 + k + koff + 1];
    acc = wmma_f32(a, b, acc);
  }
  return acc;
}

static __device__ __forceinline__ void tstore(float* O, int ldo, int n0, v8f acc) {
  const int lane = threadIdx.x & 31;
  const int l15  = lane & 15;
  const int m0   = (lane >> 4) * 8;
#pragma unroll
  for (int i = 0; i < 8; ++i) O[(m0 + i) * ldo + n0 + l15] = acc[i];
}

static __device__ __forceinline__ float sig(float x) { return 1.f / (1.f + expf(-x)); }
static __device__ __forceinline__ float lrelu(float x) { return x >= 0.f ? x : 0.01f * x; }
static __device__ __forceinline__ void atomic_add_f32(float* p, float v) { unsafeAtomicAdd(p, v); }

// ---------------- tiny const precompute: a = egv_lv1@ve_w ; b = egv_lv2@a ; c1 = ls_v@|a|
__global__ void setup_consts_kernel(const float* __restrict__ egv_lv1,
                                    const float* __restrict__ egv_lv2,
                                    const float* __restrict__ egv_ls,
                                    const float* __restrict__ ve_w,
                                    float* __restrict__ c_b, float* __restrict__ c_c1) {
  __shared__ float a[64], absa[64];
  int t = threadIdx.x;
  const float* lv1 = egv_lv1 + (size_t)L5 * 64 * 64;
  const float* lv2 = egv_lv2 + (size_t)L5 * 64 * 64;
  const float* ls  = egv_ls  + (size_t)L5 * 64 * 128;
  const float* vw  = ve_w + L5 * 64;
  float s = 0.f;
  for (int c = 0; c < 64; ++c) s += lv1[t * 64 + c] * vw[c];
  a[t] = s; absa[t] = fabsf(s);
  __syncthreads();
  float bb = 0.f, cc = 0.f;
  for (int h = 0; h < 64; ++h) { bb += lv2[t * 64 + h] * a[h]; cc += ls[t * 128 + h] * absa[h]; }
  c_b[t] = bb; c_c1[t] = cc;
}

__global__ void zero_kernel(float* __restrict__ p, int n) {
  int i = blockIdx.x * 256 + threadIdx.x;
  if (i < n) p[i] = 0.f;
}

// ---------------- node GVLinear (gv4 index i): sca_out = out_sca, vec_out = gate*out_vec
__global__ __launch_bounds__(256) void node_gv4_kernel(
    const float* __restrict__ sca_in, const float* __restrict__ vec_in,
    const float* __restrict__ lv1, const float* __restrict__ lv2,
    const float* __restrict__ ls,  const float* __restrict__ gw,
    const float* __restrict__ gbias,
    float* __restrict__ sca_out, float* __restrict__ vec_out) {
  __shared__ float S_vec[3][16][NV_];   // input vec -> vi
  __shared__ float S_vn[16][NV_];       // vnorm -> gate
  __shared__ float S_sca[16][NS_];
  __shared__ float S_osca[16][NS_];
  const int tid = threadIdx.x, wave = tid >> 5;
  const int lane = tid & 31, l15 = lane & 15, m0 = (lane >> 4) * 8;
  const int nb = blockIdx.x * 16;

  for (int i = tid; i < 16 * NS_; i += 256) {
    int e = i >> 8, c = i & 255, n = nb + e;
    S_sca[e][c] = (n < NN_c) ? sca_in[(size_t)n * NS_ + c] : 0.f;
  }
  for (int i = tid; i < 16 * 192; i += 256) {
    int e = i / 192, rc = i % 192, h = rc / 3, d = rc % 3, n = nb + e;
    S_vec[d][e][h] = (n < NN_c) ? vec_in[((size_t)n * NV_ + h) * 3 + d] : 0.f;
  }
  __syncthreads();
  // vi = vec @ lv1^T (12 jobs, regs until barrier, then overwrite S_vec)
  v8f va = {}, vb = {};
  { int j = wave, d = j >> 2, n0 = (j & 3) * 16;
    va = gemm_lds(&S_vec[d][0][0], NV_, lv1, NV_, n0, NV_, va);
    if (wave < 4) { int j2 = wave + 8, d2 = j2 >> 2, n02 = (j2 & 3) * 16;
      vb = gemm_lds(&S_vec[d2][0][0], NV_, lv1, NV_, n02, NV_, vb); } }
  __syncthreads();
  { int j = wave, d = j >> 2, n0 = (j & 3) * 16;
    tstore(&S_vec[d][0][0], NV_, n0, va);
    if (wave < 4) { int j2 = wave + 8, d2 = j2 >> 2, n02 = (j2 & 3) * 16;
      tstore(&S_vec[d2][0][0], NV_, n02, vb); } }
  __syncthreads();
  for (int i = tid; i < 16 * NV_; i += 256) {
    int e = i >> 6, h = i & 63; float s = 1e-12f;
#pragma unroll
    for (int d = 0; d < 3; ++d) { float x = S_vec[d][e][h]; s += x * x; }
    S_vn[e][h] = sqrtf(s);
  }
  __syncthreads();
  for (int rep = 0; rep < 2; ++rep) {
    int j = wave + rep * 8, n0 = j * 16; v8f acc = {};
    acc = gemm_cat320(&S_vn[0][0], &S_sca[0][0], ls, n0, acc);
    tstore(&S_osca[0][0], NS_, n0, acc);
  }
  __syncthreads();
  // gate (jobs 0..3) + out_vec (jobs 4..15, regs)
  v8f ova = {}, ovb = {};
  int j0 = wave, j1 = wave + 8;
  if (j0 < 4) {
    int n0 = j0 * 16; v8f acc = {};
    acc = gemm_lds(&S_osca[0][0], NS_, gw, NS_, n0, NS_, acc);
#pragma unroll
    for (int i = 0; i < 8; ++i) { int r = m0 + i, cn = n0 + l15;
      S_vn[r][cn] = sig(acc[i] + gbias[cn]); }
  } else {
    int jj = j0 - 4, d = jj >> 2, n0 = (jj & 3) * 16;
    ova = gemm_lds(&S_vec[d][0][0], NV_, lv2, NV_, n0, NV_, ova);
  }
  { int jj = j1 - 4, d = jj >> 2, n0 = (jj & 3) * 16;
    ovb = gemm_lds(&S_vec[d][0][0], NV_, lv2, NV_, n0, NV_, ovb); }
  __syncthreads();
  for (int i = tid; i < 16 * NS_; i += 256) {
    int e = i >> 8, c = i & 255, n = nb + e;
    if (n < NN_c) sca_out[(size_t)n * NS_ + c] = S_osca[e][c];
  }
  if (j0 >= 4) {
    int jj = j0 - 4, d = jj >> 2, n0 = (jj & 3) * 16;
#pragma unroll
    for (int i = 0; i < 8; ++i) { int e = m0 + i, h = n0 + l15, n = nb + e;
      if (n < NN_c) vec_out[((size_t)n * NV_ + h) * 3 + d] = S_vn[e][h] * ova[i]; }
  }
  { int jj = j1 - 4, d = jj >> 2, n0 = (jj & 3) * 16;
#pragma unroll
    for (int i = 0; i < 8; ++i) { int e = m0 + i, h = n0 + l15, n = nb + e;
      if (n < NN_c) vec_out[((size_t)n * NV_ + h) * 3 + d] = S_vn[e][h] * ovb[i]; }
  }
}

// ---------------- fused edge/message kernel: 16 edges per block
__global__ __launch_bounds__(256) void edge_kernel(
    const float* __restrict__ pos, const float* __restrict__ edge_feature,
    const int* __restrict__ edge_index,
    const float* __restrict__ nsca, const float* __restrict__ nvec,
    const float* __restrict__ cb, const float* __restrict__ cc1,
    const float* __restrict__ egv_ls64,                       // +64 col offset, ldw=128
    const float* __restrict__ egv_gw, const float* __restrict__ egv_gb,
    const float* __restrict__ egv_dir,
    const float* __restrict__ sca_w, const float* __restrict__ sca_b,
    const float* __restrict__ e2n_w, const float* __restrict__ e2n_b,
    const float* __restrict__ n2e_w, const float* __restrict__ n2e_b,
    const float* __restrict__ evn_w,
    const float* __restrict__ lv1, const float* __restrict__ lv2,
    const float* __restrict__ ls,  const float* __restrict__ gw,
    const float* __restrict__ gbias,
    float* __restrict__ asca, float* __restrict__ avec) {
  __shared__ float B1[16][64], B2[16][64], B3[16][64], B4[16][64];
  __shared__ float S_ysca[16][NS_];
  __shared__ float S_osca[16][NS_];        // cols 0..63 double as q-scratch early
  __shared__ float S_t[3][16][64];         // t -> vi -> (read by out_vec)
  __shared__ float su[16][3], susq[16], sabsu[16], sC[16], sdist[16];
  __shared__ int srow[16], scol[16];
  const int tid = threadIdx.x, wave = tid >> 5;
  const int lane = tid & 31, l15 = lane & 15, m0 = (lane >> 4) * 8;
  const int eb = blockIdx.x * 16;

  if (tid < 16) {
    int ge = eb + tid;
    int r = edge_index[ge], c = edge_index[EE_c + ge];
    srow[tid] = r; scol[tid] = c;
    float ex = pos[r * 3 + 0] - pos[c * 3 + 0];
    float ey = pos[r * 3 + 1] - pos[c * 3 + 1];
    float ez = pos[r * 3 + 2] - pos[c * 3 + 2];
    float d2 = ex * ex + ey * ey + ez * ez;
    float dist = sqrtf(d2 + 1e-12f);
    sdist[tid] = dist;
    float inv = 1.f / (dist + 1e-7f);
    float ux = ex * inv, uy = ey * inv, uz = ez * inv;
    su[tid][0] = ux; su[tid][1] = uy; su[tid][2] = uz;
    float usq = ux * ux + uy * uy + uz * uz;
    susq[tid] = usq; sabsu[tid] = sqrtf(usq);
    float Cc = 0.5f * (cosf(dist * (3.14159265358979f / 10.f)) + 1.f);
    sC[tid] = (dist <= 10.f) ? Cc : 0.f;
  }
  __syncthreads();
  const float GSTEP = 10.f / 59.f;
  const float GCOEF = -0.5f / (GSTEP * GSTEP);
  for (int i = tid; i < 16 * 64; i += 256) {       // esca -> B1
    int e = i >> 6, c = i & 63; float v;
    if (c < 60) { float df = sdist[e] - (float)c * GSTEP; v = expf(GCOEF * df * df); }
    else v = edge_feature[(size_t)(eb + e) * 4 + (c - 60)];
    B1[e][c] = v;
  }
  __syncthreads();
  // E1: es_pre = esca @ ls_s^T + |u|*c1  -> B2
  if (wave < 4) {
    int n0 = wave * 16; v8f acc = {};
    acc = gemm_lds(&B1[0][0], 64, egv_ls64, 128, n0, 64, acc);
#pragma unroll
    for (int i = 0; i < 8; ++i) { int r = m0 + i, cn = n0 + l15;
      B2[r][cn] = acc[i] + sabsu[r] * cc1[cn]; }
  }
  __syncthreads();
  // E2: g2 = sigmoid(es_pre@gw^T+gb)*b -> B4 (waves0-3) ; es = lrelu(es_pre) -> B3 (waves4-7)
  if (wave < 4) {
    int n0 = wave * 16; v8f acc = {};
    acc = gemm_lds(&B2[0][0], 64, egv_gw, 64, n0, 64, acc);
#pragma unroll
    for (int i = 0; i < 8; ++i) { int r = m0 + i, cn = n0 + l15;
      B4[r][cn] = sig(acc[i] + egv_gb[cn]) * cb[cn]; }
  } else {
    for (int i = tid - 128; i < 16 * 64; i += 128) {
      int e = i >> 6, c = i & 63; B3[e][c] = lrelu(B2[e][c]); }
  }
  __syncthreads();
  // E3: dd2 = g2 @ dir^T -> B1 ; then VN-leaky scalar algebra -> h in B1
  if (wave < 4) {
    int n0 = wave * 16; v8f acc = {};
    acc = gemm_lds(&B4[0][0], 64, egv_dir, 64, n0, 64, acc);
    tstore(&B1[0][0], 64, n0, acc);
  }
  __syncthreads();
  for (int i = tid; i < 16 * 64; i += 256) {
    int e = i >> 6, o = i & 63;
    float g = B4[e][o], dd = B1[e][o], uq = susq[e];
    float dot = g * dd * uq, dsq = dd * dd * uq;
    float hp = g - (dot / (dsq + 1e-6f)) * dd;
    B1[e][o] = 0.01f * g + 0.99f * ((dot >= 0.f) ? g : hp);
  }
  __syncthreads();
  // E4: 28 jobs: ysca(16), q(4), A1(4), p(4)
  for (int rep = 0; rep < 4; ++rep) {
    int j = wave + rep * 8;
    if (j >= 28) continue;
    v8f acc = {};
    if (j < 16) {
      int n0 = j * 16;
      acc = gemm_lds(&B3[0][0], 64, sca_w, 64, n0, 64, acc);
#pragma unroll
      for (int i = 0; i < 8; ++i) { int r = m0 + i, cn = n0 + l15;
        S_ysca[r][cn] = (acc[i] + sca_b[cn]) * nsca[(size_t)scol[r] * NS_ + cn]; }
    } else if (j < 20) {
      int n0 = (j - 16) * 16;
      acc = gemm_lds(&B1[0][0], 64, evn_w, 64, n0, 64, acc);
#pragma unroll
      for (int i = 0; i < 8; ++i) { int r = m0 + i, cn = n0 + l15; S_osca[r][cn] = acc[i]; }
    } else if (j < 24) {
      int n0 = (j - 20) * 16;
      acc = gemm_lds(&B3[0][0], 64, e2n_w, 64, n0, 64, acc);
#pragma unroll
      for (int i = 0; i < 8; ++i) { int r = m0 + i, cn = n0 + l15; B4[r][cn] = acc[i] + e2n_b[cn]; }
    } else {
      int n0 = (j - 24) * 16;
      const size_t abase = (size_t)scol[l15] * NS_;
      const int koff = (lane >> 4) << 1;
      const float* wrow = n2e_w + (size_t)(n0 + l15) * NS_;
      for (int k = 0; k < NS_; k += 4) {
        v2f a, b;
        a.x = nsca[abase + k + koff]; a.y = nsca[abase + k + koff + 1];
        b.x = wrow[k + koff];         b.y = wrow[k + koff + 1];
        acc = wmma_f32(a, b, acc);
      }
#pragma unroll
      for (int i = 0; i < 8; ++i) { int r = m0 + i, cn = n0 + l15; B2[r][cn] = acc[i] + n2e_b[cn]; }
    }
  }
  __syncthreads();
  // r = p*q -> B2 ; t[d] = A1 * nvec_j -> S_t
  for (int i = tid; i < 16 * 64; i += 256) {
    int e = i >> 6, c = i & 63; B2[e][c] = B2[e][c] * S_osca[e][c]; }
  for (int i = tid; i < 3 * 16 * 64; i += 256) {
    int d = i >> 10, rem = i & 1023, e = rem >> 6, c = rem & 63;
    S_t[d][e][c] = B4[e][c] * nvec[((size_t)scol[e] * NV_ + c) * 3 + d]; }
  __syncthreads();
  // E5: vi = t @ lv1^T (12 jobs) ; lr = r @ lv1^T (4 jobs) ; regs until barrier
  v8f e5a = {}, e5b = {};
  { int j = wave, d = j >> 2, n0 = (j & 3) * 16;
    e5a = gemm_lds(&S_t[d][0][0], 64, lv1, 64, n0, 64, e5a);
    int j2 = wave + 8;
    if (j2 < 12) { int d2 = j2 >> 2, n02 = (j2 & 3) * 16;
      e5b = gemm_lds(&S_t[d2][0][0], 64, lv1, 64, n02, 64, e5b); }
    else { int n02 = (j2 - 12) * 16;
      e5b = gemm_lds(&B2[0][0], 64, lv1, 64, n02, 64, e5b); } }
  __syncthreads();
  { int j = wave, d = j >> 2, n0 = (j & 3) * 16;
    tstore(&S_t[d][0][0], 64, n0, e5a);
    int j2 = wave + 8;
    if (j2 < 12) { int d2 = j2 >> 2, n02 = (j2 & 3) * 16; tstore(&S_t[d2][0][0], 64, n02, e5b); }
    else { int n02 = (j2 - 12) * 16; tstore(&B4[0][0], 64, n02, e5b); } }
  __syncthreads();
  // vi += lr*u ; vnorm -> B1
  for (int i = tid; i < 16 * 64; i += 256) {
    int e = i >> 6, h = i & 63;
    float lr = B4[e][h]; float s = 1e-12f;
#pragma unroll
    for (int d = 0; d < 3; ++d) { float v = S_t[d][e][h] + lr * su[e][d]; S_t[d][e][h] = v; s += v * v; }
    B1[e][h] = sqrtf(s);
  }
  __syncthreads();
  // E6: out_sca = [vnorm | y_sca] @ ls^T  (16 jobs, K=320)
  for (int rep = 0; rep < 2; ++rep) {
    int j = wave + rep * 8, n0 = j * 16; v8f acc = {};
    acc = gemm_cat320(&B1[0][0], &S_ysca[0][0], ls, n0, acc);
    tstore(&S_osca[0][0], NS_, n0, acc);
  }
  __syncthreads();
  // E7: gate (0..3) -> B4 ; out_vec (4..15, regs)
  v8f ova = {}, ovb = {};
  if (wave < 4) {
    int n0 = wave * 16; v8f acc = {};
    acc = gemm_lds(&S_osca[0][0], NS_, gw, NS_, n0, NS_, acc);
#pragma unroll
    for (int i = 0; i < 8; ++i) { int r = m0 + i, cn = n0 + l15;
      B4[r][cn] = sig(acc[i] + gbias[cn]); }
  } else {
    int jj = wave - 4, d = jj >> 2, n0 = (jj & 3) * 16;
    ova = gemm_lds(&S_t[d][0][0], 64, lv2, 64, n0, 64, ova);
  }
  { int jj = wave + 4, d = jj >> 2, n0 = (jj & 3) * 16;
    ovb = gemm_lds(&S_t[d][0][0], 64, lv2, 64, n0, 64, ovb); }
  __syncthreads();
  // E8: cutoff + atomic scatter to destination nodes (row)
  for (int i = tid; i < 16 * NS_; i += 256) {
    int e = i >> 8, c = i & 255;
    atomic_add_f32(&asca[(size_t)srow[e] * NS_ + c], S_osca[e][c] * sC[e]);
  }
  if (wave >= 4) {
    int jj = wave - 4, d = jj >> 2, n0 = (jj & 3) * 16;
#pragma unroll
    for (int i = 0; i < 8; ++i) { int e = m0 + i, h = n0 + l15;
      atomic_add_f32(&avec[((size_t)srow[e] * NV_ + h) * 3 + d], B4[e][h] * ova[i] * sC[e]); }
  }
  { int jj = wave + 4, d = jj >> 2, n0 = (jj & 3) * 16;
#pragma unroll
    for (int i = 0; i < 8; ++i) { int e = m0 + i, h = n0 + l15;
      atomic_add_f32(&avec[((size_t)srow[e] * NV_ + h) * 3 + d], B4[e][h] * ovb[i] * sC[e]); }
  }
}

// ---------------- node post: LN + VN-leaky + gv4(3) + residual -> d_out
__global__ __launch_bounds__(256) void node_post_kernel(
    const float* __restrict__ node_sca, const float* __restrict__ node_vec,
    const float* __restrict__ csca, const float* __restrict__ cvec,
    const float* __restrict__ asca, const float* __restrict__ avec,
    const float* __restrict__ ln_sg, const float* __restrict__ ln_sb,
    const float* __restrict__ ln_vg, const float* __restrict__ ln_vb,
    const float* __restrict__ blkdir,
    const float* __restrict__ lv1, const float* __restrict__ lv2,
    const float* __restrict__ ls,  const float* __restrict__ gw,
    const float* __restrict__ gbias,
    float* __restrict__ out) {
  __shared__ float S_osca[16][NS_];   // osca -> leaky(osca)
  __shared__ float S_o2[16][NS_];     // dsca
  __shared__ float S_x[3][16][NV_];   // ovec -> vnl result
  __shared__ float S_d[3][16][NV_];   // blkdir@ovec -> vi
  __shared__ float S_vn[16][NV_];     // vnorm -> gate
  const int tid = threadIdx.x, wave = tid >> 5;
  const int lane = tid & 31, l15 = lane & 15, m0 = (lane >> 4) * 8;
  const int nb = blockIdx.x * 16;
  const size_t VOFF = (size_t)NN_c * NS_;

  for (int i = tid; i < 16 * NS_; i += 256) {
    int e = i >> 8, c = i & 255, n = nb + e;
    S_osca[e][c] = (n < NN_c) ? csca[(size_t)n * NS_ + c] + asca[(size_t)n * NS_ + c] : 0.f;
  }
  for (int i = tid; i < 16 * 192; i += 256) {
    int e = i / 192, rc = i % 192, h = rc / 3, d = rc % 3, n = nb + e;
    size_t idx = ((size_t)n * NV_ + h) * 3 + d;
    S_x[d][e][h] = (n < NN_c) ? cvec[idx] + avec[idx] : 0.f;
  }
  __syncthreads();
  if (tid < 16) {               // LayerNorm sca (axis -1)
    int e = tid; float m = 0.f;
    for (int c = 0; c < NS_; ++c) m += S_osca[e][c];
    m *= (1.f / NS_);
    float v = 0.f;
    for (int c = 0; c < NS_; ++c) { float d0 = S_osca[e][c] - m; v += d0 * d0; }
    v *= (1.f / NS_);
    float inv = rsqrtf(v + 1e-5f);
    for (int c = 0; c < NS_; ++c) S_osca[e][c] = (S_osca[e][c] - m) * inv * ln_sg[c] + ln_sb[c];
  } else if (tid < 32) {        // LayerNorm vec (axes -2,-1)
    int e = tid - 16; float m = 0.f;
    for (int h = 0; h < NV_; ++h) for (int d = 0; d < 3; ++d) m += S_x[d][e][h];
    m *= (1.f / 192.f);
    float v = 0.f;
    for (int h = 0; h < NV_; ++h) for (int d = 0; d < 3; ++d) { float d0 = S_x[d][e][h] - m; v += d0 * d0; }
    v *= (1.f / 192.f);
    float inv = rsqrtf(v + 1e-5f);
    for (int h = 0; h < NV_; ++h) for (int d = 0; d < 3; ++d)
      S_x[d][e][h] = (S_x[d][e][h] - m) * inv * ln_vg[h * 3 + d] + ln_vb[h * 3 + d];
  }
  __syncthreads();
  // d = blkdir @ ovec (12 jobs) -> S_d
  { int j = wave, d = j >> 2, n0 = (j & 3) * 16; v8f acc = {};
    acc = gemm_lds(&S_x[d][0][0], NV_, blkdir, NV_, n0, NV_, acc);
    tstore(&S_d[d][0][0], NV_, n0, acc);
    if (wave < 4) { int j2 = wave + 8, d2 = j2 >> 2, n02 = (j2 & 3) * 16; v8f a2 = {};
      a2 = gemm_lds(&S_x[d2][0][0], NV_, blkdir, NV_, n02, NV_, a2);
      tstore(&S_d[d2][0][0], NV_, n02, a2); } }
  __syncthreads();
  // VN leaky relu (full form) -> S_x ; leaky on S_osca
  for (int i = tid; i < 16 * NV_; i += 256) {
    int e = i >> 6, o = i & 63;
    float dot = 0.f, dsq = 0.f;
#pragma unroll
    for (int d = 0; d < 3; ++d) { float x = S_x[d][e][o], dd = S_d[d][e][o]; dot += x * dd; dsq += dd * dd; }
    float coef = dot / (dsq + 1e-6f);
    bool neg = dot < 0.f;
#pragma unroll
    for (int d = 0; d < 3; ++d) { float x = S_x[d][e][o], dd = S_d[d][e][o];
      float xp = x - coef * dd;
      S_x[d][e][o] = 0.01f * x + 0.99f * (neg ? xp : x); }
  }
  for (int i = tid; i < 16 * NS_; i += 256) {
    int e = i >> 8, c = i & 255; S_osca[e][c] = lrelu(S_osca[e][c]); }
  __syncthreads();
  // vi = x @ lv1^T (12 jobs) -> S_d (dead)
  { int j = wave, d = j >> 2, n0 = (j & 3) * 16; v8f acc = {};
    acc = gemm_lds(&S_x[d][0][0], NV_, lv1, NV_, n0, NV_, acc);
    tstore(&S_d[d][0][0], NV_, n0, acc);
    if (wave < 4) { int j2 = wave + 8, d2 = j2 >> 2, n02 = (j2 & 3) * 16; v8f a2 = {};
      a2 = gemm_lds(&S_x[d2][0][0], NV_, lv1, NV_, n02, NV_, a2);
      tstore(&S_d[d2][0][0], NV_, n02, a2); } }
  __syncthreads();
  for (int i = tid; i < 16 * NV_; i += 256) {
    int e = i >> 6, h = i & 63; float s = 1e-12f;
#pragma unroll
    for (int d = 0; d < 3; ++d) { float x = S_d[d][e][h]; s += x * x; }
    S_vn[e][h] = sqrtf(s);
  }
  __syncthreads();
  // dsca = [vnorm | leaky(osca)] @ ls^T (16 jobs) -> S_o2
  for (int rep = 0; rep < 2; ++rep) {
    int j = wave + rep * 8, n0 = j * 16; v8f acc = {};
    acc = gemm_cat320(&S_vn[0][0], &S_osca[0][0], ls, n0, acc);
    tstore(&S_o2[0][0], NS_, n0, acc);
  }
  __syncthreads();
  // gate (0..3) -> S_vn ; out_vec (4..15, regs)
  v8f ova = {}, ovb = {};
  int j0 = wave, j1 = wave + 8;
  if (j0 < 4) {
    int n0 = j0 * 16; v8f acc = {};
    acc = gemm_lds(&S_o2[0][0], NS_, gw, NS_, n0, NS_, acc);
#pragma unroll
    for (int i = 0; i < 8; ++i) { int r = m0 + i, cn = n0 + l15;
      S_vn[r][cn] = sig(acc[i] + gbias[cn]); }
  } else {
    int jj = j0 - 4, d = jj >> 2, n0 = (jj & 3) * 16;
    ova = gemm_lds(&S_d[d][0][0], NV_, lv2, NV_, n0, NV_, ova);
  }
  { int jj = j1 - 4, d = jj >> 2, n0 = (jj & 3) * 16;
    ovb = gemm_lds(&S_d[d][0][0], NV_, lv2, NV_, n0, NV_, ovb); }
  __syncthreads();
  // residual stores
  for (int i = tid; i < 16 * NS_; i += 256) {
    int e = i >> 8, c = i & 255, n = nb + e;
    if (n < NN_c) out[(size_t)n * NS_ + c] = node_sca[(size_t)n * NS_ + c] + S_o2[e][c];
  }
  if (j0 >= 4) {
    int jj = j0 - 4, d = jj >> 2, n0 = (jj & 3) * 16;
#pragma unroll
    for (int i = 0; i < 8; ++i) { int e = m0 + i, h = n0 + l15, n = nb + e;
      if (n < NN_c) { size_t idx = ((size_t)n * NV_ + h) * 3 + d;
        out[VOFF + idx] = node_vec[idx] + S_vn[e][h] * ova[i]; } }
  }
  { int jj = j1 - 4, d = jj >> 2, n0 = (jj & 3) * 16;
#pragma unroll
    for (int i = 0; i < 8; ++i) { int e = m0 + i, h = n0 + l15, n = nb + e;
      if (n < NN_c) { size_t idx = ((size_t)n * NV_ + h) * 3 + d;
        out[VOFF + idx] = node_vec[idx] + S_vn[e][h] * ovb[i]; } }
  }
}

extern "C" void kernel_launch(void* const* d_in, const int* in_sizes, int n_in,
                              void* d_out, int out_size, void* d_ws, size_t ws_size,
                              hipStream_t stream) {
  const float* node_sca = (const float*)d_in[0];
  const float* node_vec = (const float*)d_in[1];
  const float* pos = (const float*)d_in[2];
  const float* edge_feature = (const float*)d_in[3];
  const int*   edge_index = (const int*)d_in[4];
  const float* gv4_lv1 = (const float*)d_in[6];
  const float* gv4_lv2 = (const float*)d_in[7];
  const float* gv4_ls  = (const float*)d_in[8];
  const float* gv4_gw  = (const float*)d_in[9];
  const float* gv4_gb  = (const float*)d_in[10];
  const float* egv_lv1 = (const float*)d_in[11];
  const float* egv_lv2 = (const float*)d_in[12];
  const float* egv_ls  = (const float*)d_in[13];
  const float* egv_gw  = (const float*)d_in[14];
  const float* egv_gb  = (const float*)d_in[15];
  const float* egv_dir = (const float*)d_in[16];
  const float* blk_dir = (const float*)d_in[17];
  const float* sca_w = (const float*)d_in[18];
  const float* sca_b = (const float*)d_in[19];
  const float* e2n_w = (const float*)d_in[20];
  const float* e2n_b = (const float*)d_in[21];
  const float* n2e_w = (const float*)d_in[22];
  const float* n2e_b = (const float*)d_in[23];
  const float* evn_w = (const float*)d_in[24];
  const float* ve_w  = (const float*)d_in[25];
  const float* ln_sg = (const float*)d_in[26];
  const float* ln_sb = (const float*)d_in[27];
  const float* ln_vg = (const float*)d_in[28];
  const float* ln_vb = (const float*)d_in[29];

  float* ws = (float*)d_ws;
  float* w_nsca = ws;
  float* w_nvec = w_nsca + (size_t)NN_c * NS_;
  float* w_asca = w_nvec + (size_t)NN_c * NV_ * 3;
  float* w_avec = w_asca + (size_t)NN_c * NS_;
  float* w_cb   = w_avec + (size_t)NN_c * NV_ * 3;
  float* w_cc1  = w_cb + 64;

  // gv4 weight slices (layer 5, indices 0..3)
  auto LV1 = [&](int i) { return gv4_lv1 + (size_t)(L5 * 4 + i) * 64 * 64; };
  auto LV2 = [&](int i) { return gv4_lv2 + (size_t)(L5 * 4 + i) * 64 * 64; };
  auto LSW = [&](int i) { return gv4_ls  + (size_t)(L5 * 4 + i) * 256 * 320; };
  auto GW  = [&](int i) { return gv4_gw  + (size_t)(L5 * 4 + i) * 64 * 256; };
  auto GB  = [&](int i) { return gv4_gb  + (size_t)(L5 * 4 + i) * 64; };

  setup_consts_kernel<<<1, 64, 0, stream>>>(egv_lv1, egv_lv2, egv_ls, ve_w, w_cb, w_cc1);

  int nz = NN_c * (NS_ + NV_ * 3);
  zero_kernel<<<(nz + 255) / 256, 256, 0, stream>>>(w_asca, nz);

  int nblk = (NN_c + 15) / 16;
  node_gv4_kernel<<<nblk, 256, 0, stream>>>(node_sca, node_vec,
      LV1(0), LV2(0), LSW(0), GW(0), GB(0), w_nsca, w_nvec);

  edge_kernel<<<EE_c / 16, 256, 0, stream>>>(pos, edge_feature, edge_index,
      w_nsca, w_nvec, w_cb, w_cc1,
      egv_ls + (size_t)L5 * 64 * 128 + 64,
      egv_gw + (size_t)L5 * 64 * 64, egv_gb + L5 * 64,
      egv_dir + (size_t)L5 * 64 * 64,
      sca_w + (size_t)L5 * 256 * 64, sca_b + L5 * 256,
      e2n_w + (size_t)L5 * 64 * 64, e2n_b + L5 * 64,
      n2e_w + (size_t)L5 * 64 * 256, n2e_b + L5 * 64,
      evn_w + (size_t)L5 * 64 * 64,
      LV1(1), LV2(1), LSW(1), GW(1), GB(1),
      w_asca, w_avec);

  node_gv4_kernel<<<nblk, 256, 0, stream>>>(node_sca, node_vec,
      LV1(2), LV2(2), LSW(2), GW(2), GB(2), w_nsca, w_nvec);

  node_post_kernel<<<nblk, 256, 0, stream>>>(node_sca, node_vec,
      w_nsca, w_nvec, w_asca, w_avec,
      ln_sg + L5 * 256, ln_sb + L5 * 256, ln_vg + L5 * 192, ln_vb + L5 * 192,
      blk_dir + (size_t)L5 * 64 * 64,
      LV1(3), LV2(3), LSW(3), GW(3), GB(3),
      (float*)d_out);
}